// PartialSoftmaxDistiller_6141803233758
// MI455X (gfx1250) — compile-verified
//
#include <hip/hip_runtime.h>

// PartialSoftmaxDistiller, MI455X / gfx1250 (wave32).
//
// Closed-form refactor of the reference:
//   per sample i (negatives only): Et = sum exp(t), Es = sum exp(s),
//                                  A  = sum exp(t)*(t - s)
//   per positive j:  Zt = Et + exp(t_j); Zs = Es + exp(s_j)
//                    KL_j = (A + exp(t_j)*(t_j - s_j))/Zt + log(Zs) - log(Zt)
//   loss = sum_{i,j} KL_j / N
//
// Cross-lane reductions run through V_WMMA_F32_16X16X4_F32 with an all-ones
// B matrix (layout-independent for B): one WMMA folds lanes (l, l+16), the
// 8-entry D accumulator sum folds 8 M-rows, one shfl_xor(16) finishes.

#define N_SAMP 64
#define N_CLS  1024

typedef __attribute__((ext_vector_type(2))) float v2f;
typedef __attribute__((ext_vector_type(8))) float v8f;

// Full wave32 fp32 sum; EXEC must be all ones at the call site.
__device__ __forceinline__ float wave_sum_wmma(float v) {
    v2f a; a[0] = v;    a[1] = 0.0f;   // A[m][0]=v(lane m), A[m][2]=v(lane m+16)
    v2f b; b[0] = 1.0f; b[1] = 1.0f;   // ones: D[m][n] = sum_k A[m][k]
    v8f c = {};
    c = __builtin_amdgcn_wmma_f32_16x16x4_f32(
            /*neg_a=*/false, a, /*neg_b=*/false, b,
            /*c_mod=*/(short)0, c, /*reuse_a=*/false, /*reuse_b=*/false);
    // Lane n<16 holds D[0..7][n]; lane n+16 holds D[8..15][n].
    float r = ((c[0] + c[1]) + (c[2] + c[3])) + ((c[4] + c[5]) + (c[6] + c[7]));
    r += __shfl_xor(r, 16, 32);        // combine the two M-halves
    return r;                           // every lane holds the wave sum
}

__global__ __launch_bounds__(256)
void psd_sample_kernel(const float* __restrict__ student,
                       const float* __restrict__ teacher,
                       const int*   __restrict__ target,
                       float*       __restrict__ partial) {
    const int i    = blockIdx.x;        // sample
    const int tid  = threadIdx.x;       // 256 threads, 8 waves
    const int lane = tid & 31;
    const int wave = tid >> 5;
    const int base = i * N_CLS + tid * 4;

    const float4 sv = *reinterpret_cast<const float4*>(student + base);
    const float4 tv = *reinterpret_cast<const float4*>(teacher + base);
    const int4   pv = *reinterpret_cast<const int4*>(target + base);

    const float se[4] = {sv.x, sv.y, sv.z, sv.w};
    const float te[4] = {tv.x, tv.y, tv.z, tv.w};
    const int   pg[4] = {pv.x, pv.y, pv.z, pv.w};

    float ets[4], ess[4], dts[4];
    float Et = 0.0f, Es = 0.0f, A = 0.0f;
#pragma unroll
    for (int r = 0; r < 4; ++r) {
        const float d  = te[r] - se[r];
        const float et = __expf(te[r]);
        const float es = __expf(se[r]);
        ets[r] = et; ess[r] = es; dts[r] = d;
        if (pg[r] == 0) {               // negatives feed the shared sums
            Et += et; Es += es; A += et * d;
        }
    }

    // Wave-level reductions on the matrix pipe (uniform control flow here).
    Et = wave_sum_wmma(Et);
    Es = wave_sum_wmma(Es);
    A  = wave_sum_wmma(A);

    __shared__ float red[8][3];
    __shared__ float tot[3];
    if (lane == 0) { red[wave][0] = Et; red[wave][1] = Es; red[wave][2] = A; }
    __syncthreads();
    if (tid == 0) {
        float e0 = 0.0f, e1 = 0.0f, e2 = 0.0f;
#pragma unroll
        for (int w = 0; w < 8; ++w) { e0 += red[w][0]; e1 += red[w][1]; e2 += red[w][2]; }
        tot[0] = e0; tot[1] = e1; tot[2] = e2;
    }
    __syncthreads();
    Et = tot[0]; Es = tot[1]; A = tot[2];

    // Per-positive KL terms.
    float kl = 0.0f;
#pragma unroll
    for (int r = 0; r < 4; ++r) {
        if (pg[r] == 1) {
            const float Zt = Et + ets[r];
            const float Zs = Es + ess[r];
            kl += (A + ets[r] * dts[r]) / Zt + __logf(Zs) - __logf(Zt);
        }
    }
    kl = wave_sum_wmma(kl);             // call site is convergent

    __shared__ float kred[8];
    if (lane == 0) kred[wave] = kl;
    __syncthreads();
    if (tid == 0) {
        float acc = 0.0f;
#pragma unroll
        for (int w = 0; w < 8; ++w) acc += kred[w];
        partial[i] = acc;               // deterministic per-block partial
    }
}

__global__ __launch_bounds__(32)
void psd_finalize_kernel(const float* __restrict__ partial,
                         float*       __restrict__ out) {
    const int lane = threadIdx.x;       // exactly one wave32
    const float v = partial[lane] + partial[lane + 32];
    const float r = wave_sum_wmma(v);
    if (lane == 0) out[0] = r * (1.0f / (float)N_SAMP);
}

extern "C" void kernel_launch(void* const* d_in, const int* in_sizes, int n_in,
                              void* d_out, int out_size, void* d_ws, size_t ws_size,
                              hipStream_t stream) {
    (void)in_sizes; (void)n_in; (void)out_size; (void)ws_size;
    const float* student = (const float*)d_in[0];
    const float* teacher = (const float*)d_in[1];
    const int*   target  = (const int*)d_in[2];
    float* partial = (float*)d_ws;      // 64 floats of scratch
    float* out     = (float*)d_out;     // scalar loss

    psd_sample_kernel<<<N_SAMP, 256, 0, stream>>>(student, teacher, target, partial);
    psd_finalize_kernel<<<1, 32, 0, stream>>>(partial, out);
}